// MambaBlock_9689446220504
// MI455X (gfx1250) — compile-verified
//
#include <hip/hip_runtime.h>
#include <hip/hip_bf16.h>

typedef __attribute__((ext_vector_type(2))) float v2f;
typedef __attribute__((ext_vector_type(4))) float v4f;
typedef __attribute__((ext_vector_type(8))) float v8f;

#define B_SZ   4
#define L_SZ   2048
#define DMODEL 1024
#define DSTATE 16
#define DCONV  4
#define DINNER 2048
#define NROWS  (B_SZ * L_SZ)   // 8192 token rows

__device__ __forceinline__ float sigmoidf_(float v) { return 1.0f / (1.0f + expf(-v)); }
__device__ __forceinline__ float siluf_(float v)    { return v * sigmoidf_(v); }

// ---------------------------------------------------------------------------
// fp32 GEMM via V_WMMA_F32_16X16X4_F32.  C[M,N] = A[M,K] * B[K,N], row-major.
// Workgroup = 256 threads (8 waves) -> 128x128 C tile; wave -> 32x64 (2x4
// WMMA fragments, 64 acc VGPRs).  Double-buffered LDS ping-pong: stage s+1
// global loads are issued before stage-s WMMAs, one barrier per stage.
// B tile is stored TRANSPOSED in LDS so both fragment elements (K, K+1) are
// contiguous -> single ds_load_b64 per fragment, stride-34 conflict-free.
// Requires M%128==0, N%128==0, K%32==0 (true for all calls here).
// ---------------------------------------------------------------------------
#define TILE_M 128
#define TILE_N 128
#define TILE_K 32
#define LDT    (TILE_K + 2)   // 34: conflict-free 16-lane column reads, 8B pairs

__global__ __launch_bounds__(256) void gemm_f32_wmma(
    const float* __restrict__ A, const float* __restrict__ Bm,
    float* __restrict__ C, int M, int N, int K)
{
  __shared__ float As[2][TILE_M][LDT];   // [stage][m][k]
  __shared__ float Bt[2][TILE_N][LDT];   // [stage][n][k]  (transposed fill)

  const int tid  = threadIdx.x;
  const int lane = tid & 31;
  const int wave = tid >> 5;     // 0..7
  const int wm   = wave >> 1;    // 0..3 : 32-row band
  const int wn   = wave & 1;     // 0..1 : 64-col band
  const int hi   = lane >> 4;    // half-wave: K pair select / M+8 on store
  const int lo   = lane & 15;

  const int tileM = blockIdx.y * TILE_M;
  const int tileN = blockIdx.x * TILE_N;

  v8f acc[2][4] = {};

  // A fill: 128 rows x 32 cols, float4; 8 thr/row, 32 rows/pass, 4 passes
  const int arow = tid >> 3;            // 0..31
  const int acol = (tid & 7) << 2;      // 0,4,...,28
  // B fill: 32 rows x 128 cols, float4; 32 thr/row, 8 rows/pass, 4 passes
  const int brow = tid >> 5;            // 0..7
  const int bcol = (tid & 31) << 2;     // 0,4,...,124

  const int S = K / TILE_K;
  v4f aReg[4], bReg[4];

  // ---- prologue: stage 0 ----
#pragma unroll
  for (int p = 0; p < 4; ++p) {
    aReg[p] = *(const v4f*)(A + (size_t)(tileM + arow + p * 32) * K + acol);
    bReg[p] = *(const v4f*)(Bm + (size_t)(brow + p * 8) * N + (tileN + bcol));
  }
#pragma unroll
  for (int p = 0; p < 4; ++p) {
    const int r = arow + p * 32;
    As[0][r][acol + 0] = aReg[p].x; As[0][r][acol + 1] = aReg[p].y;
    As[0][r][acol + 2] = aReg[p].z; As[0][r][acol + 3] = aReg[p].w;
    const int kr = brow + p * 8;
    Bt[0][bcol + 0][kr] = bReg[p].x;
    Bt[0][bcol + 1][kr] = bReg[p].y;
    Bt[0][bcol + 2][kr] = bReg[p].z;
    Bt[0][bcol + 3][kr] = bReg[p].w;
  }
  __syncthreads();

  for (int s = 0; s < S; ++s) {
    const int cur = s & 1;
    const int nxt = cur ^ 1;

    // prefetch stage s+1 while stage s computes (latency overlap)
    if (s + 1 < S) {
      const int kb = (s + 1) * TILE_K;
#pragma unroll
      for (int p = 0; p < 4; ++p) {
        aReg[p] = *(const v4f*)(A + (size_t)(tileM + arow + p * 32) * K + (kb + acol));
        bReg[p] = *(const v4f*)(Bm + (size_t)(kb + brow + p * 8) * N + (tileN + bcol));
      }
    }

#pragma unroll
    for (int k0 = 0; k0 < TILE_K; k0 += 4) {
      // ISA layout: A/B vgpr j -> K=k0+j (lanes 0-15) / K=k0+j+2 (lanes 16-31)
      const int kk = k0 + (hi << 1);
      v2f af[2], bf[4];
#pragma unroll
      for (int i = 0; i < 2; ++i) {
        const int row = wm * 32 + i * 16 + lo;
        af[i].x = As[cur][row][kk];
        af[i].y = As[cur][row][kk + 1];
      }
#pragma unroll
      for (int j = 0; j < 4; ++j) {
        const int col = wn * 64 + j * 16 + lo;
        bf[j].x = Bt[cur][col][kk];
        bf[j].y = Bt[cur][col][kk + 1];
      }
#pragma unroll
      for (int i = 0; i < 2; ++i)
#pragma unroll
        for (int j = 0; j < 4; ++j)
          acc[i][j] = __builtin_amdgcn_wmma_f32_16x16x4_f32(
              false, af[i], false, bf[j], (short)0, acc[i][j], false, false);
    }

    if (s + 1 < S) {
      // ping-pong: buffer `nxt` was last READ in stage s-1; every thread passed
      // the stage-(s-1) barrier after those reads, so filling it now is safe.
#pragma unroll
      for (int p = 0; p < 4; ++p) {
        const int r = arow + p * 32;
        As[nxt][r][acol + 0] = aReg[p].x; As[nxt][r][acol + 1] = aReg[p].y;
        As[nxt][r][acol + 2] = aReg[p].z; As[nxt][r][acol + 3] = aReg[p].w;
        const int kr = brow + p * 8;
        Bt[nxt][bcol + 0][kr] = bReg[p].x;
        Bt[nxt][bcol + 1][kr] = bReg[p].y;
        Bt[nxt][bcol + 2][kr] = bReg[p].z;
        Bt[nxt][bcol + 3][kr] = bReg[p].w;
      }
      __syncthreads();   // single barrier per stage
    }
  }

  // C/D layout: vgpr v -> M = v (lanes 0-15) / v+8 (lanes 16-31), N = lane%16
#pragma unroll
  for (int i = 0; i < 2; ++i) {
#pragma unroll
    for (int j = 0; j < 4; ++j) {
      const int col = tileN + wn * 64 + j * 16 + lo;
#pragma unroll
      for (int v = 0; v < 8; ++v) {
        const int row = tileM + wm * 32 + i * 16 + v + (hi << 3);
        C[(size_t)row * N + col] = acc[i][j][v];
      }
    }
  }
}

// ---------------------------------------------------------------------------
// Depthwise causal conv (width 4) + bias + SiLU over x_ssm = xz[..., :DINNER]
// ---------------------------------------------------------------------------
__global__ __launch_bounds__(256) void conv_silu_kernel(
    const float* __restrict__ xz, const float* __restrict__ conv_w,
    const float* __restrict__ conv_b, float* __restrict__ xc)
{
  const size_t idx = (size_t)blockIdx.x * 256 + threadIdx.x;   // over NROWS*DINNER
  const int d = (int)(idx % DINNER);
  const int l = (int)((idx / DINNER) % L_SZ);
  const int b = (int)(idx / ((size_t)DINNER * L_SZ));
  float acc = conv_b[d];
#pragma unroll
  for (int k = 0; k < DCONV; ++k) {
    const int lk = l + k - (DCONV - 1);
    if (lk >= 0)
      acc += xz[((size_t)b * L_SZ + lk) * (2 * DINNER) + d] * conv_w[d * DCONV + k];
  }
  xc[idx] = siluf_(acc);
}

// ---------------------------------------------------------------------------
// x_dbl = xc @ W_x  (K=2048, N=33).  One block per token row; strided K
// partials in registers, wave shuffle-reduce, cross-wave via LDS.
// ---------------------------------------------------------------------------
__global__ __launch_bounds__(256) void xdbl_kernel(
    const float* __restrict__ xc, const float* __restrict__ Wx,
    float* __restrict__ dbl)
{
  const int row  = blockIdx.x;
  const int tid  = threadIdx.x;
  const int lane = tid & 31;
  const int wave = tid >> 5;

  float acc[33];
#pragma unroll
  for (int j = 0; j < 33; ++j) acc[j] = 0.0f;

  const float* xr = xc + (size_t)row * DINNER;
  for (int k = tid; k < DINNER; k += 256) {
    const float xv = xr[k];
    const float* w = Wx + (size_t)k * 33;
#pragma unroll
    for (int j = 0; j < 33; ++j) acc[j] += xv * w[j];
  }
#pragma unroll
  for (int off = 16; off > 0; off >>= 1)
#pragma unroll
    for (int j = 0; j < 33; ++j) acc[j] += __shfl_down(acc[j], off, 32);

  __shared__ float red[8][33];
  if (lane == 0) {
#pragma unroll
    for (int j = 0; j < 33; ++j) red[wave][j] = acc[j];
  }
  __syncthreads();
  if (tid < 33) {
    float s = 0.0f;
#pragma unroll
    for (int w = 0; w < 8; ++w) s += red[w][tid];
    dbl[(size_t)row * 33 + tid] = s;
  }
}

// ---------------------------------------------------------------------------
// Per-(row, n): P = [ A_bar = exp(softplus(dt)*(-exp(A_log))) | dt*B | C ]
// ---------------------------------------------------------------------------
__global__ __launch_bounds__(256) void param_kernel(
    const float* __restrict__ dbl, const float* __restrict__ A_log,
    float* __restrict__ P)
{
  const int idx = blockIdx.x * 256 + threadIdx.x;   // NROWS*DSTATE
  const int n   = idx & (DSTATE - 1);
  const int row = idx >> 4;
  const float x0    = dbl[(size_t)row * 33];
  const float delta = (x0 > 20.0f) ? x0 : log1pf(expf(x0));
  const float Bv    = dbl[(size_t)row * 33 + 1 + n];
  const float Cv    = dbl[(size_t)row * 33 + 1 + DSTATE + n];
  const float Aa    = -expf(A_log[n]);
  P[(size_t)row * 48 + n]          = expf(delta * Aa);
  P[(size_t)row * 48 + DSTATE + n] = delta * Bv;
  P[(size_t)row * 48 + 32 + n]     = Cv;
}

// ---------------------------------------------------------------------------
// Selective scan + D skip + SiLU(z) gate.  Grid = B * (DINNER/512); each
// thread owns 2 channels' h[16] in registers.  Params double-buffered in LDS.
// ---------------------------------------------------------------------------
__global__ __launch_bounds__(256) void scan_kernel(
    const float* __restrict__ P, const float* __restrict__ xc,
    const float* __restrict__ xz, const float* __restrict__ D_par,
    float* __restrict__ y2)
{
  const int tid   = threadIdx.x;
  const int b     = blockIdx.x >> 2;
  const int chunk = blockIdx.x & 3;
  const int d0    = chunk * 512 + tid;          // first owned channel
  const int d1    = d0 + 256;                   // second owned channel

  __shared__ float ld[2][48];

  const float* Pb  = P  + (size_t)b * L_SZ * 48;
  const float* xcb = xc + (size_t)b * L_SZ * DINNER;
  const float* zb  = xz + (size_t)b * L_SZ * (2 * DINNER) + DINNER;
  float*       yb  = y2 + (size_t)b * L_SZ * DINNER;

  const float Dp0 = D_par[d0];
  const float Dp1 = D_par[d1];

  float h0[DSTATE], h1[DSTATE];
#pragma unroll
  for (int n = 0; n < DSTATE; ++n) { h0[n] = 0.0f; h1[n] = 0.0f; }

  for (int l = 0; l < L_SZ; ++l) {
    const int cur = l & 1;
    if (tid < 48) ld[cur][tid] = Pb[(size_t)l * 48 + tid];
    __syncthreads();   // double buffer: one barrier/step is sufficient

    const float x0 = xcb[(size_t)l * DINNER + d0];
    const float x1 = xcb[(size_t)l * DINNER + d1];
    float y0 = 0.0f, y1 = 0.0f;
#pragma unroll
    for (int n = 0; n < DSTATE; ++n) {
      const float a  = ld[cur][n];
      const float bb = ld[cur][DSTATE + n];
      const float c  = ld[cur][32 + n];
      h0[n] = a * h0[n] + bb * x0;
      h1[n] = a * h1[n] + bb * x1;
      y0 += h0[n] * c;
      y1 += h1[n] * c;
    }
    y0 += x0 * Dp0;
    y1 += x1 * Dp1;
    const float z0 = zb[(size_t)l * (2 * DINNER) + d0];
    const float z1 = zb[(size_t)l * (2 * DINNER) + d1];
    yb[(size_t)l * DINNER + d0] = y0 * siluf_(z0);
    yb[(size_t)l * DINNER + d1] = y1 * siluf_(z1);
  }
}

// ---------------------------------------------------------------------------
extern "C" void kernel_launch(void* const* d_in, const int* in_sizes, int n_in,
                              void* d_out, int out_size, void* d_ws, size_t ws_size,
                              hipStream_t stream)
{
  const float* x      = (const float*)d_in[0];
  const float* W_in   = (const float*)d_in[1];
  const float* conv_w = (const float*)d_in[2];
  const float* conv_b = (const float*)d_in[3];
  const float* W_x    = (const float*)d_in[4];
  const float* A_log  = (const float*)d_in[5];
  const float* D_par  = (const float*)d_in[6];
  const float* W_out  = (const float*)d_in[7];
  float* out = (float*)d_out;

  char* ws = (char*)d_ws;
  size_t off = 0;
  float* xz  = (float*)(ws + off); off += (size_t)NROWS * 2 * DINNER * sizeof(float); // 128 MB
  float* xc  = (float*)(ws + off); off += (size_t)NROWS * DINNER * sizeof(float);     // 64 MB
  float* dbl = (float*)(ws + off); off += (size_t)NROWS * 33 * sizeof(float);
  float* P   = (float*)(ws + off); off += (size_t)NROWS * 48 * sizeof(float);
  float* y2  = (float*)(ws + off);                                                    // 64 MB

  // 1) xz = x @ W_in          (8192x1024 * 1024x4096)
  gemm_f32_wmma<<<dim3((2 * DINNER) / TILE_N, NROWS / TILE_M), 256, 0, stream>>>(
      x, W_in, xz, NROWS, 2 * DINNER, DMODEL);

  // 2) xc = silu(causal depthwise conv(x_ssm) + b)
  conv_silu_kernel<<<(NROWS * DINNER) / 256, 256, 0, stream>>>(xz, conv_w, conv_b, xc);

  // 3) x_dbl = xc @ W_x       (K=2048, N=33)
  xdbl_kernel<<<NROWS, 256, 0, stream>>>(xc, W_x, dbl);

  // 4) P = [A_bar | delta*B | C]
  param_kernel<<<(NROWS * DSTATE) / 256, 256, 0, stream>>>(dbl, A_log, P);

  // 5) selective scan + D skip + silu(z) gate  -> y2
  scan_kernel<<<B_SZ * (DINNER / 512), 256, 0, stream>>>(P, xc, xz, D_par, y2);

  // 6) out = y2 @ W_out       (8192x2048 * 2048x1024)
  gemm_f32_wmma<<<dim3(DMODEL / TILE_N, NROWS / TILE_M), 256, 0, stream>>>(
      y2, W_out, out, NROWS, DMODEL, DINNER);
}